// MVG_binaryNet_29283087024616
// MI455X (gfx1250) — compile-verified
//
#include <hip/hip_runtime.h>
#include <hip/hip_bf16.h>

// MVG binaryNet forward on gfx1250 (MI455X).
// All three batched 256^3 GEMMs run on V_WMMA_F32_16X16X4_F32 (exact fp32).
// xcov2[b] (256KB) lives in the 320KB WGP LDS and is consumed in place by
// the layer-3 double GEMM; sigma3 lands in d_out's xcov3 slot and is
// finalized in place.  Register blocking: phase1 uses 2x2 tiles per wave
// (4 WMMA / 4 fragment loads), phase2 uses 2 tiles per wave sharing the B
// fragment (2 WMMA / 3 fragment loads).

typedef float v2f __attribute__((ext_vector_type(2)));
typedef float v8f __attribute__((ext_vector_type(8)));

#define SQ2PI 0.79788456f

__device__ __forceinline__ v8f wmma_f32(v2f a, v2f b, v8f c) {
    return __builtin_amdgcn_wmma_f32_16x16x4_f32(
        /*neg_a=*/false, a, /*neg_b=*/false, b,
        /*c_mod=*/(short)0, c, /*reuse_a=*/false, /*reuse_b=*/false);
}

// ---------------- small elementwise / reduction kernels ----------------

__global__ void k_sigmoid(const float* __restrict__ w, float* __restrict__ m, int n) {
    int i = blockIdx.x * blockDim.x + threadIdx.x;
    if (i < n) {
        float s = 1.0f / (1.0f + expf(-w[i]));
        m[i] = 2.0f * s - 1.0f;
    }
}

__global__ void k_square(const float* __restrict__ a, float* __restrict__ b, int n) {
    int i = blockIdx.x * blockDim.x + threadIdx.x;
    if (i < n) b[i] = a[i] * a[i];
}

// at[c*R + r] = a[r*C + c]
__global__ void k_transpose(const float* __restrict__ a, float* __restrict__ at,
                            int R, int C) {
    int i = blockIdx.x * 256 + threadIdx.x;
    if (i < R * C) {
        int r = i / C, c = i % C;
        at[c * R + r] = a[i];
    }
}

// s[i] = sum_j (1 - m[j,i]^2), m is [rows,256] row-major. 1 block x 256.
__global__ void k_colsum(const float* __restrict__ m, float* __restrict__ s, int rows) {
    int i = threadIdx.x;
    float acc = 0.0f;
    for (int j = 0; j < rows; ++j) {
        float v = m[j * 256 + i];
        acc += 1.0f - v * v;
    }
    s[i] = acc;
}

// ---------------- layer 1: h1 = x @ m0 + th0 ; activations ----------------
// grid 32 x 256 threads (8 waves/block, one 16x16 tile per wave).
__global__ void k_layer1(const float* __restrict__ x, const float* __restrict__ m0T,
                         const float* __restrict__ th0, const float* __restrict__ s0,
                         float* __restrict__ x1bar, float* __restrict__ d1) {
    const int lane = threadIdx.x & 31, wave = threadIdx.x >> 5;
    const int r16 = lane & 15, hi = lane >> 4;
    const int tile = blockIdx.x * 8 + wave;
    const int it = tile >> 4, lt = tile & 15;

    v8f acc = {};
    const float* arow = x + (it * 16 + r16) * 784 + hi * 2;
    const float* brow = m0T + (lt * 16 + r16) * 784 + hi * 2;
    for (int k0 = 0; k0 < 784; k0 += 4) {
        v2f a = *(const v2f*)(arow + k0);
        v2f b = *(const v2f*)(brow + k0);
        acc = wmma_f32(a, b, acc);
    }
    int col = lt * 16 + r16;
    float inv = rsqrtf(s0[col]);
    float th = th0[col];
#pragma unroll
    for (int v = 0; v < 8; ++v) {
        int row = it * 16 + v + 8 * hi;
        float h = acc[v] + th;
        float xb = tanhf(SQ2PI * h * inv);
        x1bar[row * 256 + col] = xb;
        d1[row * 256 + col] = 1.0f - xb * xb;
    }
}

// ---------------- layer 2 vectors: hbar2 = x1@m1, diagsig2 = d1@m1sq -----
__global__ void k_layer2(const float* __restrict__ x1bar, const float* __restrict__ d1,
                         const float* __restrict__ m1T, const float* __restrict__ m1sqT,
                         const float* __restrict__ th1, const float* __restrict__ s1,
                         float* __restrict__ x2bar, float* __restrict__ c2,
                         float* __restrict__ d2) {
    const int lane = threadIdx.x & 31, wave = threadIdx.x >> 5;
    const int r16 = lane & 15, hi = lane >> 4;
    const int tile = blockIdx.x * 8 + wave;
    const int it = tile >> 4, lt = tile & 15;

    v8f acc1 = {}, acc2 = {};
    const float* a1row = x1bar + (it * 16 + r16) * 256 + hi * 2;
    const float* a2row = d1 + (it * 16 + r16) * 256 + hi * 2;
    const float* b1row = m1T + (lt * 16 + r16) * 256 + hi * 2;
    const float* b2row = m1sqT + (lt * 16 + r16) * 256 + hi * 2;
    for (int k0 = 0; k0 < 256; k0 += 4) {
        v2f a1 = *(const v2f*)(a1row + k0);
        v2f b1 = *(const v2f*)(b1row + k0);
        v2f a2 = *(const v2f*)(a2row + k0);
        v2f b2 = *(const v2f*)(b2row + k0);
        acc1 = wmma_f32(a1, b1, acc1);
        acc2 = wmma_f32(a2, b2, acc2);
    }
    int col = lt * 16 + r16;
    float th = th1[col];
    float sc = s1[col];
#pragma unroll
    for (int v = 0; v < 8; ++v) {
        int row = it * 16 + v + 8 * hi;
        float h = acc1[v] + th;
        float ds = acc2[v] + sc;
        float inv = rsqrtf(ds);
        float xb = tanhf(SQ2PI * h * inv);
        float dd = 1.0f - xb * xb;
        x2bar[row * 256 + col] = xb;
        c2[row * 256 + col] = dd * inv;
        d2[row * 256 + col] = dd;
    }
}

// ---------------- fused per-batch sigma3 kernel ----------------
// One workgroup (256 thr / 8 waves) per batch b; ~296KB LDS -> 1 WG per WGP.
__global__ void k_sigma3(const float* __restrict__ m1T, const float* __restrict__ m2T,
                         const float* __restrict__ d1, const float* __restrict__ c2,
                         const float* __restrict__ d2, const float* __restrict__ s1,
                         float* __restrict__ sigma3out) {
    extern __shared__ float lds[];
    const int XS = 260;                    // X row stride (260 % 64 = 4 banks)
    const int TS = 258;                    // strip row stride (258 % 64 = 2)
    float* X    = lds;                     // 256*260
    float* m2sT = lds + 256 * XS;          // 16 x 258  (strip of m2, transposed)
    float* tstT = m2sT + 16 * TS;          // 16 x 258  (t-strip, transposed)
    float* d1b  = tstT + 16 * TS;          // 256
    float* c2b  = d1b + 256;               // 256
    float* d2b  = c2b + 256;               // 256
    float* s1v  = d2b + 256;               // 256

    const int b = blockIdx.x;
    const int t = threadIdx.x;
    d1b[t] = d1[b * 256 + t];
    c2b[t] = c2[b * 256 + t];
    d2b[t] = d2[b * 256 + t];
    s1v[t] = s1[t];
    __syncthreads();

    const int lane = t & 31, wave = t >> 5;
    const int r16 = lane & 15, hi = lane >> 4;

    // ---- Phase 1: X[i,l] = SQ2PI*(G2[i,l] + (i==l)s1)*c2[i]c2[l] + (i==l)d2[i]
    // 2x2 tile blocking: 8x8 macro-tiles (32x32 each) over 8 waves.
    for (int mt = wave; mt < 64; mt += 8) {
        int mi = mt >> 3, ml = mt & 7;
        v8f c00 = {}, c01 = {}, c10 = {}, c11 = {};
        const float* arow0 = m1T + (mi * 32 + r16) * 256 + hi * 2;
        const float* arow1 = arow0 + 16 * 256;
        const float* brow0 = m1T + (ml * 32 + r16) * 256 + hi * 2;
        const float* brow1 = brow0 + 16 * 256;
        const float* dptr = d1b + hi * 2;
        for (int k0 = 0; k0 < 256; k0 += 4) {
            v2f dv = *(const v2f*)(dptr + k0);
            v2f a0 = *(const v2f*)(arow0 + k0);
            v2f a1 = *(const v2f*)(arow1 + k0);
            a0.x *= dv.x; a0.y *= dv.y;            // A[i,j] = m1[j,i]*d1[b,j]
            a1.x *= dv.x; a1.y *= dv.y;
            v2f b0 = *(const v2f*)(brow0 + k0);
            v2f b1 = *(const v2f*)(brow1 + k0);
            c00 = wmma_f32(a0, b0, c00);
            c01 = wmma_f32(a0, b1, c01);
            c10 = wmma_f32(a1, b0, c10);
            c11 = wmma_f32(a1, b1, c11);
        }
#pragma unroll
        for (int ii = 0; ii < 2; ++ii) {
#pragma unroll
            for (int ll = 0; ll < 2; ++ll) {
                v8f acc = (ii == 0) ? (ll == 0 ? c00 : c01) : (ll == 0 ? c10 : c11);
                int l = ml * 32 + ll * 16 + r16;
                float cl = c2b[l];
#pragma unroll
                for (int v = 0; v < 8; ++v) {
                    int i = mi * 32 + ii * 16 + v + 8 * hi;
                    float val = acc[v];
                    if (i == l) val += s1v[i];
                    val = SQ2PI * val * c2b[i] * cl;
                    if (i == l) val += d2b[i];
                    X[i * XS + l] = val;
                }
            }
        }
    }
    __syncthreads();

    // ---- Phase 2: per 16-column strip of m2; each wave owns 2 tiles
    for (int lt = 0; lt < 16; ++lt) {
        // stage m2 strip transposed: m2sT[l][k] = m2[k, lt*16+l] = m2T[lt*16+l][k]
        for (int idx = t; idx < 4096; idx += 256) {
            int l = idx >> 8, k = idx & 255;
            m2sT[l * TS + k] = m2T[(lt * 16 + l) * 256 + k];
        }
        __syncthreads();

        // 2a: t[j,l] = sum_k X[j,k] * m2[k, lt*16+l]; store transposed tstT[l][j]
        {
            int jt0 = wave, jt1 = wave + 8;
            v8f acc0 = {}, acc1 = {};
            const float* arow0 = X + (jt0 * 16 + r16) * XS + hi * 2;
            const float* arow1 = X + (jt1 * 16 + r16) * XS + hi * 2;
            const float* brow = m2sT + r16 * TS + hi * 2;
            for (int k0 = 0; k0 < 256; k0 += 4) {
                v2f bb = *(const v2f*)(brow + k0);
                v2f a0 = *(const v2f*)(arow0 + k0);
                v2f a1 = *(const v2f*)(arow1 + k0);
                acc0 = wmma_f32(a0, bb, acc0);
                acc1 = wmma_f32(a1, bb, acc1);
            }
#pragma unroll
            for (int v = 0; v < 8; ++v) {
                tstT[r16 * TS + jt0 * 16 + v + 8 * hi] = acc0[v];
                tstT[r16 * TS + jt1 * 16 + v + 8 * hi] = acc1[v];
            }
        }
        __syncthreads();

        // 2b: sigma3[i, lt*16+l] = sum_j m2T[i,j] * t[j,l]  (raw, no diag(s2))
        {
            int it0 = wave, it1 = wave + 8;
            v8f acc0 = {}, acc1 = {};
            const float* arow0 = m2T + (it0 * 16 + r16) * 256 + hi * 2;
            const float* arow1 = m2T + (it1 * 16 + r16) * 256 + hi * 2;
            const float* brow = tstT + r16 * TS + hi * 2;
            for (int k0 = 0; k0 < 256; k0 += 4) {
                v2f bb = *(const v2f*)(brow + k0);
                v2f a0 = *(const v2f*)(arow0 + k0);
                v2f a1 = *(const v2f*)(arow1 + k0);
                acc0 = wmma_f32(a0, bb, acc0);
                acc1 = wmma_f32(a1, bb, acc1);
            }
            float* orow = sigma3out + (size_t)b * 65536 + lt * 16 + r16;
#pragma unroll
            for (int v = 0; v < 8; ++v) {
                orow[(it0 * 16 + v + 8 * hi) * 256] = acc0[v];
                orow[(it1 * 16 + v + 8 * hi) * 256] = acc1[v];
            }
        }
        __syncthreads();   // protect m2sT/tstT before next strip
    }
}

// ---------------- layer 3 vectors from sigma3 diagonal ----------------
__global__ void k_layer3vec(const float* __restrict__ sigma3, const float* __restrict__ x2bar,
                            const float* __restrict__ m2, const float* __restrict__ th2,
                            const float* __restrict__ s2,
                            float* __restrict__ x3bar, float* __restrict__ c3,
                            float* __restrict__ d3) {
    int b = blockIdx.x, i = threadIdx.x;
    float diag = sigma3[(size_t)b * 65536 + i * 257] + s2[i];
    float acc = 0.0f;
    for (int j = 0; j < 256; ++j)
        acc += x2bar[b * 256 + j] * m2[j * 256 + i];
    float h = acc + th2[i];
    float inv = rsqrtf(diag);
    float xb = tanhf(SQ2PI * h * inv);
    float dd = 1.0f - xb * xb;
    x3bar[b * 256 + i] = xb;
    c3[b * 256 + i] = dd * inv;
    d3[b * 256 + i] = dd;
}

// ---------------- finalize xcov3 in place in d_out ----------------
__global__ void k_finalize(float* __restrict__ xcov3, const float* __restrict__ c3,
                           const float* __restrict__ d3, const float* __restrict__ s2) {
    int idx = blockIdx.x * 256 + threadIdx.x;   // grid 65536 -> 16777216 elems
    int b = idx >> 16, rem = idx & 65535;
    int i = rem >> 8, l = rem & 255;
    float val = xcov3[idx];
    if (i == l) val += s2[i];
    val = SQ2PI * val * c3[b * 256 + i] * c3[b * 256 + l];
    if (i == l) val += d3[b * 256 + i];
    xcov3[idx] = val;
}

// ---------------- head: logits, log_softmax, loss, accuracy ----------------
__global__ void k_head(const float* __restrict__ x3bar, const float* __restrict__ mlast,
                       const float* __restrict__ thlast, const int* __restrict__ target,
                       float* __restrict__ out) {
    __shared__ float redl[256];
    __shared__ float redc[256];
    int b = threadIdx.x;
    float h[10];
#pragma unroll
    for (int o = 0; o < 10; ++o) h[o] = thlast[o];
    for (int j = 0; j < 256; ++j) {
        float xv = x3bar[b * 256 + j];
#pragma unroll
        for (int o = 0; o < 10; ++o) h[o] += xv * mlast[j * 10 + o];
    }
    float mx = h[0]; int am = 0;
#pragma unroll
    for (int o = 1; o < 10; ++o) { if (h[o] > mx) { mx = h[o]; am = o; } }
    float se = 0.0f;
#pragma unroll
    for (int o = 0; o < 10; ++o) se += expf(h[o] - mx);
    float lse = mx + logf(se);
    int tg = target[b];
    float lossi = 0.0f;
#pragma unroll
    for (int o = 0; o < 10; ++o) {
        out[b * 10 + o] = h[o];
        float lp = h[o] - lse;
        out[2560 + b * 10 + o] = lp;
        if (o == tg) lossi = -lp;
    }
    redl[b] = lossi;
    redc[b] = (am == tg) ? 1.0f : 0.0f;
    __syncthreads();
    for (int s = 128; s > 0; s >>= 1) {
        if (b < s) { redl[b] += redl[b + s]; redc[b] += redc[b + s]; }
        __syncthreads();
    }
    if (b == 0) {
        out[16782336] = redl[0] * (1.0f / 256.0f);
        out[16782337] = redc[0] * (1.0f / 256.0f);
    }
}

extern "C" void kernel_launch(void* const* d_in, const int* in_sizes, int n_in,
                              void* d_out, int out_size, void* d_ws, size_t ws_size,
                              hipStream_t stream) {
    const float* x      = (const float*)d_in[0];
    const int*   target = (const int*)  d_in[1];
    const float* w0     = (const float*)d_in[2];
    const float* w1     = (const float*)d_in[3];
    const float* w2     = (const float*)d_in[4];
    const float* wlast  = (const float*)d_in[5];
    const float* th0    = (const float*)d_in[6];
    const float* th1    = (const float*)d_in[7];
    const float* th2    = (const float*)d_in[8];
    const float* thlast = (const float*)d_in[9];
    float* out = (float*)d_out;

    // workspace carve-up (~5 MB of fp32)
    float* p = (float*)d_ws;
    float* m0     = p; p += 784 * 256;
    float* m0T    = p; p += 784 * 256;
    float* m1     = p; p += 65536;
    float* m2     = p; p += 65536;
    float* mlast  = p; p += 2560;
    float* m1T    = p; p += 65536;
    float* m1sqT  = p; p += 65536;
    float* m2T    = p; p += 65536;
    float* s0     = p; p += 256;
    float* s1     = p; p += 256;
    float* s2     = p; p += 256;
    float* x1     = p; p += 65536;
    float* d1v    = p; p += 65536;
    float* x2     = p; p += 65536;
    float* c2v    = p; p += 65536;
    float* d2v    = p; p += 65536;
    float* x3     = p; p += 65536;
    float* c3v    = p; p += 65536;
    float* d3v    = p; p += 65536;

    float* sigma3 = out + 5120;   // xcov3 slot of d_out, first holds raw sigma3

    k_sigmoid<<<(784 * 256 + 255) / 256, 256, 0, stream>>>(w0, m0, 784 * 256);
    k_sigmoid<<<256, 256, 0, stream>>>(w1, m1, 65536);
    k_sigmoid<<<256, 256, 0, stream>>>(w2, m2, 65536);
    k_sigmoid<<<10, 256, 0, stream>>>(wlast, mlast, 2560);

    k_transpose<<<(784 * 256 + 255) / 256, 256, 0, stream>>>(m0, m0T, 784, 256);
    k_transpose<<<256, 256, 0, stream>>>(m1, m1T, 256, 256);
    k_transpose<<<256, 256, 0, stream>>>(m2, m2T, 256, 256);
    k_square  <<<256, 256, 0, stream>>>(m1T, m1sqT, 65536);

    k_colsum<<<1, 256, 0, stream>>>(m0, s0, 784);
    k_colsum<<<1, 256, 0, stream>>>(m1, s1, 256);
    k_colsum<<<1, 256, 0, stream>>>(m2, s2, 256);

    k_layer1<<<32, 256, 0, stream>>>(x, m0T, th0, s0, x1, d1v);
    k_layer2<<<32, 256, 0, stream>>>(x1, d1v, m1T, m1sqT, th1, s1, x2, c2v, d2v);

    size_t ldsBytes = (size_t)(256 * 260 + 2 * 16 * 258 + 1024) * sizeof(float); // ~296 KB
    k_sigma3<<<256, 256, ldsBytes, stream>>>(m1T, m2T, d1v, c2v, d2v, s1, sigma3);

    k_layer3vec<<<256, 256, 0, stream>>>(sigma3, x2, m2, th2, s2, x3, c3v, d3v);
    k_finalize <<<65536, 256, 0, stream>>>(sigma3, c3v, d3v, s2);
    k_head     <<<1, 256, 0, stream>>>(x3, mlast, thlast, target, out);
}